// AUAvULoss_15573551415397
// MI455X (gfx1250) — compile-verified
//
#include <hip/hip_runtime.h>
#include <math.h>

namespace {
constexpr int   kNTh      = 21;
constexpr int   kHistSize = 4 * kNTh;   // 84 counters: [ac, au, ic, iu] x 21 bins
constexpr float kEps      = 1e-10f;
}

typedef float v2f __attribute__((ext_vector_type(2)));
typedef float v8f __attribute__((ext_vector_type(8)));

// ---- CDNA5 async global->LDS path (gfx1250), with safe fallback -------------
#if defined(__HIP_DEVICE_COMPILE__)
#if __has_builtin(__builtin_amdgcn_global_load_async_to_lds_b128) && \
    __has_builtin(__builtin_amdgcn_s_wait_asynccnt)
#define USE_ASYNC_LDS 1
#endif
#endif
#ifndef USE_ASYNC_LDS
#define USE_ASYNC_LDS 0
#endif

#if USE_ASYNC_LDS
typedef int v4i __attribute__((ext_vector_type(4)));
typedef __attribute__((address_space(1))) v4i gas_v4i;  // global (AS1) int4
typedef __attribute__((address_space(3))) v4i las_v4i;  // LDS    (AS3) int4

__device__ __forceinline__ void async_copy_b128(void* lds_dst, const void* gsrc) {
  __builtin_amdgcn_global_load_async_to_lds_b128(
      (gas_v4i*)gsrc, (las_v4i*)lds_dst, 0, 0);
}
#endif

// ---- Kernel 0: workspace init (idempotent; runs every launch) ---------------
__global__ void avu_init_kernel(unsigned* __restrict__ mm, float* __restrict__ ghist) {
  const int t = threadIdx.x;
  if (t == 0) mm[0] = 0x7F800000u;  // +inf bits (min accumulator)
  if (t == 1) mm[1] = 0x00000000u;  // 0.0f bits (max accumulator; unc >= 0)
  for (int i = t; i < kHistSize; i += blockDim.x) ghist[i] = 0.0f;
}

// ---- Kernel 1: min/max of unc (async-LDS streamed, wave32 reduce) -----------
__global__ void avu_minmax_kernel(const float* __restrict__ unc,
                                  unsigned* __restrict__ mm, int n4) {
  const float4* u4 = (const float4*)unc;
  const int tid = threadIdx.x;
  const int gstride = gridDim.x * blockDim.x;
  int i = blockIdx.x * blockDim.x + tid;
  float vmin = __uint_as_float(0x7F800000u);
  float vmax = 0.0f;

#if USE_ASYNC_LDS
  __shared__ float4 stage[2][256];
  if (i < n4) async_copy_b128(&stage[0][tid], &u4[i]);
  int buf = 0;
  for (; i < n4; i += gstride, buf ^= 1) {
    const int inext = i + gstride;
    if (inext < n4) {
      async_copy_b128(&stage[buf ^ 1][tid], &u4[inext]);
      __builtin_amdgcn_s_wait_asynccnt(1);  // oldest copy complete (in-order)
    } else {
      __builtin_amdgcn_s_wait_asynccnt(0);
    }
    // Each lane reads only its own slot, written by its own wave's async op.
    float4 v = stage[buf][tid];
    vmin = fminf(vmin, fminf(fminf(v.x, v.y), fminf(v.z, v.w)));
    vmax = fmaxf(vmax, fmaxf(fmaxf(v.x, v.y), fmaxf(v.z, v.w)));
  }
#else
  for (; i < n4; i += gstride) {
    float4 v = u4[i];
    vmin = fminf(vmin, fminf(fminf(v.x, v.y), fminf(v.z, v.w)));
    vmax = fmaxf(vmax, fmaxf(fmaxf(v.x, v.y), fmaxf(v.z, v.w)));
  }
#endif

  for (int off = 16; off > 0; off >>= 1) {
    vmin = fminf(vmin, __shfl_xor(vmin, off, 32));
    vmax = fmaxf(vmax, __shfl_xor(vmax, off, 32));
  }
  if ((tid & 31) == 0) {
    atomicMin(&mm[0], __float_as_uint(vmin));   // nonneg floats order like uints
    atomicMax(&mm[1], __float_as_uint(vmax));
  }
}

// ---- Kernel 2: binned weighted histogram (the bandwidth-bound pass) ---------
__global__ void avu_hist_kernel(const float* __restrict__ probs,
                                const int* __restrict__ labels,
                                const float* __restrict__ unc,
                                const unsigned* __restrict__ mm,
                                float* __restrict__ ghist, int n4) {
  // lh[counter][lane]: bank = (bin*32 + lane) mod 64 -> conflict-free across a
  // wave32's lanes for ds_add_f32, regardless of bin distribution.
  __shared__ float lh[kHistSize * 32];
  const int tid  = threadIdx.x;
  const int lane = tid & 31;
  for (int i = tid; i < kHistSize * 32; i += blockDim.x) lh[i] = 0.0f;
  __syncthreads();

  const float umin  = __uint_as_float(mm[0]);
  const float umax  = __uint_as_float(mm[1]);
  const float range = umax - umin;
  const float scale = (range > 0.0f) ? (20.0f / range) : 0.0f;

  const float4* u4 = (const float4*)unc;
  const float4* p4 = (const float4*)probs;
  const int4*   l4 = (const int4*)labels;
  const int gstride = gridDim.x * blockDim.x;

  for (int q = blockIdx.x * blockDim.x + tid; q < n4; q += gstride) {
    const int qn = q + gstride;
    if (qn < n4) {  // hint next grid-stride tile (global_prefetch_b8)
      __builtin_prefetch(&u4[qn], 0, 0);
      __builtin_prefetch(&p4[2 * qn], 0, 0);
      __builtin_prefetch(&l4[qn], 0, 0);
    }
    const float4 u  = u4[q];
    const float4 pa = p4[2 * q];       // (p0,p1) of elements 4q, 4q+1
    const float4 pb = p4[2 * q + 1];   // (p0,p1) of elements 4q+2, 4q+3
    const int4   lb = l4[q];

    const float us[4]  = {u.x, u.y, u.z, u.w};
    const float p0s[4] = {pa.x, pa.z, pb.x, pb.z};
    const float p1s[4] = {pa.y, pa.w, pb.y, pb.w};
    const int   ls[4]  = {lb.x, lb.y, lb.z, lb.w};

#pragma unroll
    for (int e = 0; e < 4; ++e) {
      const float uu   = us[e];
      const float conf = p1s[e];
      const int   pred = (p1s[e] > p0s[e]) ? 1 : 0;   // argmax, tie -> 0
      const bool  acc  = (ls[e] == pred);
      const float t    = tanhf(uu);                    // native TRANS on CDNA5
      // certain at threshold T  <=>  ceil((uu-umin)*20/range) <= T
      int bin = (int)ceilf((uu - umin) * scale);
      bin = bin < 0 ? 0 : (bin > 20 ? 20 : bin);
      const float base = acc ? conf : (1.0f - conf);
      const int   cb   = acc ? 0 : 2;                  // {ac,au} vs {ic,iu}
      atomicAdd(&lh[((cb + 0) * kNTh + bin) * 32 + lane], base * (1.0f - t));
      atomicAdd(&lh[((cb + 1) * kNTh + bin) * 32 + lane], base * t);
    }
  }
  __syncthreads();

  for (int c = tid; c < kHistSize; c += blockDim.x) {
    float s = 0.0f;
#pragma unroll
    for (int l = 0; l < 32; ++l) s += lh[c * 32 + l];
    atomicAdd(&ghist[c], s);  // global_atomic_add_f32
  }
}

// ---- Kernel 3: WMMA prefix sums + AVU curve + trapezoid + loss (one wave) ---
// Prefix sums over bins are P = L x H with L = 21x21 lower-triangular ones,
// H = 21x4 histogram. Computed exactly in f32 with V_WMMA_F32_16X16X4_F32:
// 2 M-tiles x 6 accumulator-chained K-chunks. Single full wave32, uniform
// control flow around WMMA (EXEC all-ones requirement).
__global__ void __launch_bounds__(32) avu_final_kernel(const float* __restrict__ ghist,
                                                       float* __restrict__ out) {
  __shared__ float P[32][4];           // P[t][c] = sum_{b<=t} hist[c][b]
  const int l     = threadIdx.x;
  const int nidx  = l & 15;            // D/B column within tile
  const int khalf = (l < 16) ? 0 : 2;  // A/B K-pair selector per lane half

#pragma unroll
  for (int tile = 0; tile < 2; ++tile) {
    const int mrow0    = tile * 16;
    const int m_global = mrow0 + (l & 15);  // A row handled by this lane
    v8f acc = {0.f, 0.f, 0.f, 0.f, 0.f, 0.f, 0.f, 0.f};
#pragma unroll
    for (int kc = 0; kc < 6; ++kc) {        // K = 24 covers 21 bins (zero-pad)
      const int kg = kc * 4 + khalf;
      v2f a, b;
      // A (16x4, lower-triangular ones): lanes 0-15 hold K0/K1, 16-31 K2/K3
      a.x = (kg     <= m_global) ? 1.0f : 0.0f;
      a.y = (kg + 1 <= m_global) ? 1.0f : 0.0f;
      // B (4x16): column n = channel (only n<4 used), row k = bin
      b.x = (nidx < 4 && kg     < kNTh) ? ghist[nidx * kNTh + kg]     : 0.0f;
      b.y = (nidx < 4 && kg + 1 < kNTh) ? ghist[nidx * kNTh + kg + 1] : 0.0f;
      acc = __builtin_amdgcn_wmma_f32_16x16x4_f32(
          false, a, false, b, (short)0, acc, false, false);
    }
    // Scatter D tile to LDS: VGPR j holds M=j (lanes 0-15) / M=j+8 (16-31)
#pragma unroll
    for (int j = 0; j < 8; ++j) {
      const int row = mrow0 + j + ((l < 16) ? 0 : 8);
      if (nidx < 4 && row < kNTh) P[row][nidx] = acc[j];
    }
  }
  __syncthreads();  // single-wave workgroup: lowers to S_NOP, DS waits suffice

  const float T1 = P[kNTh - 1][1];   // total w_au
  const float T3 = P[kNTh - 1][3];   // total w_iu
  float n_ac = 0.f, n_au = 0.f, n_ic = 0.f, n_iu = 0.f;
  if (l < kNTh) {
    n_ac = P[l][0];
    n_au = T1 - P[l][1];             // suffix = total - prefix
    n_ic = P[l][2];
    n_iu = T3 - P[l][3];
  }
  const float avu = (n_ac + n_iu) / (n_ac + n_au + n_ic + n_iu + kEps);
  const float avu_next = __shfl_down(avu, 1, 32);
  float seg = (l < kNTh - 1) ? 0.5f * (avu + avu_next) * (1.0f / 20.0f) : 0.0f;
  for (int off = 16; off > 0; off >>= 1) seg += __shfl_xor(seg, off, 32);
  if (l == 0) {
    out[0] = -logf(seg + kEps);  // avu_loss (BETA = 1)
    out[1] = seg;                // auc_avu
  }
}

// -----------------------------------------------------------------------------
extern "C" void kernel_launch(void* const* d_in, const int* in_sizes, int n_in,
                              void* d_out, int out_size, void* d_ws, size_t ws_size,
                              hipStream_t stream) {
  (void)n_in; (void)out_size; (void)ws_size;
  const float* probs  = (const float*)d_in[0];   // (N,2) f32
  const int*   labels = (const int*)d_in[1];     // (N,)  int
  const float* unc    = (const float*)d_in[2];   // (N,)  f32
  float* out = (float*)d_out;                    // [avu_loss, auc_avu]

  unsigned* mm    = (unsigned*)d_ws;             // [min_bits, max_bits]
  float*    ghist = (float*)d_ws + 2;            // 84 floats

  const int n  = in_sizes[2];
  const int n4 = n >> 2;

  avu_init_kernel<<<1, 128, 0, stream>>>(mm, ghist);
  avu_minmax_kernel<<<512, 256, 0, stream>>>(unc, mm, n4);
  avu_hist_kernel<<<1024, 256, 0, stream>>>(probs, labels, unc, mm, ghist, n4);
  avu_final_kernel<<<1, 32, 0, stream>>>(ghist, out);
}